// QLV4_Embedding_MOD_38946763440163
// MI455X (gfx1250) — compile-verified
//
#include <hip/hip_runtime.h>
#include <stddef.h>

// ---------------------------------------------------------------------------
// QLV4 int4-emulated embedding gather, MI455X (gfx1250).
// out[tok, :] = weight[id, :] * scale[id]   (fp32, D=1024)
// Memory-bound: ~128 MiB of HBM traffic -> target is pure bandwidth.
// Uses CDNA5 async global->LDS DMA (ASYNCcnt) to double-buffer rows,
// b128 vector memory ops, and non-temporal stores for the streaming output.
// ---------------------------------------------------------------------------

typedef float v4f __attribute__((ext_vector_type(4)));
// Must match the builtin prototype's 'V4i' (GCC-style vector, NOT ext_vector).
typedef int v4i __attribute__((vector_size(16)));

#define EMB_D       1024          // floats per row
#define THREADS     256           // 8 waves; 256 * float4 == one full row
#define TPB_TOKENS  8             // tokens (rows) per block

#if defined(__HIP_DEVICE_COMPILE__) && \
    __has_builtin(__builtin_amdgcn_global_load_async_to_lds_b128)
  #define EMB_ASYNC 1
#else
  #define EMB_ASYNC 0
#endif

#if EMB_ASYNC
typedef __attribute__((address_space(1))) v4i* gv4i_ptr;   // global int4*
typedef __attribute__((address_space(3))) v4i* lv4i_ptr;   // LDS int4*

__device__ __forceinline__ void emb_async_wait_1() {
#if __has_builtin(__builtin_amdgcn_s_wait_asynccnt)
  __builtin_amdgcn_s_wait_asynccnt(1);
#else
  asm volatile("s_wait_asynccnt 1" ::: "memory");
#endif
}
__device__ __forceinline__ void emb_async_wait_0() {
#if __has_builtin(__builtin_amdgcn_s_wait_asynccnt)
  __builtin_amdgcn_s_wait_asynccnt(0);
#else
  asm volatile("s_wait_asynccnt 0" ::: "memory");
#endif
}

// Flat VA == global VA on amdgcn, so integer-casting a generic pointer to an
// AS(1) pointer is exact and avoids const/addrspace qualifier diagnostics.
__device__ __forceinline__ gv4i_ptr to_global_v4i(const float* p) {
  return (gv4i_ptr)(unsigned long long)p;
}
#endif

__global__ __launch_bounds__(THREADS)
void qlv4_embed_dequant_kernel(const int* __restrict__ ids,
                               const float* __restrict__ W,
                               const float* __restrict__ Sc,
                               float* __restrict__ out,
                               int n_tokens) {
  const int tid  = threadIdx.x;            // 0..255, owns float4 #tid of a row
  const int tok0 = blockIdx.x * TPB_TOKENS;

#if EMB_ASYNC
  __shared__ v4f buf[2][THREADS];          // 2 x 4 KiB double buffer

  if (tok0 + TPB_TOKENS <= n_tokens) {     // block-uniform: full tile
    float sc[2];

    // Prologue: kick off row for token 0 into slot 0.
    {
      const int id = ids[tok0];
      sc[0] = Sc[id];
      const float* src = W + (size_t)id * EMB_D + (tid << 2);
      __builtin_amdgcn_global_load_async_to_lds_b128(
          to_global_v4i(src), (lv4i_ptr)&buf[0][tid], 0, 0);
    }

#pragma unroll
    for (int t = 0; t < TPB_TOKENS; ++t) {
      const int slot = t & 1;
      if (t + 1 < TPB_TOKENS) {
        // Issue next row into the other slot, then wait for the current one.
        const int id = ids[tok0 + t + 1];
        sc[(t + 1) & 1] = Sc[id];
        const float* src = W + (size_t)id * EMB_D + (tid << 2);
        __builtin_amdgcn_global_load_async_to_lds_b128(
            to_global_v4i(src), (lv4i_ptr)&buf[(t + 1) & 1][tid], 0, 0);
        emb_async_wait_1();                // my oldest async (row t) is done
      } else {
        emb_async_wait_0();                // drain: last row done
      }
      __syncthreads();                     // row t visible block-wide

      v4f v = buf[slot][tid] * sc[slot];   // ds_load_b128 + v_mul_f32 x4
      __builtin_nontemporal_store(
          v, (v4f*)(out + (size_t)(tok0 + t) * EMB_D) + tid);

      __syncthreads();                     // all reads of buf[slot] done
                                           // before t+2 overwrites it
    }
    return;
  }
#endif

  // Direct path: fallback when async builtins are unavailable, and for any
  // partial tail block. b128 load -> scale -> non-temporal b128 store.
#pragma unroll
  for (int t = 0; t < TPB_TOKENS; ++t) {
    const int tok = tok0 + t;
    if (tok >= n_tokens) return;
    const int   id = ids[tok];
    const float s  = Sc[id];
    const v4f* src = (const v4f*)(W + (size_t)id * EMB_D) + tid;
    v4f v = *src;                          // RT load: repeated rows can hit L2
    v *= s;
    __builtin_nontemporal_store(
        v, (v4f*)(out + (size_t)tok * EMB_D) + tid);
  }
}

extern "C" void kernel_launch(void* const* d_in, const int* in_sizes, int n_in,
                              void* d_out, int out_size, void* d_ws, size_t ws_size,
                              hipStream_t stream) {
  const int*   ids = (const int*)d_in[0];    // [B*S] token indices
  const float* W   = (const float*)d_in[1];  // [V, 1024] int4-valued fp32
  const float* Sc  = (const float*)d_in[2];  // [V, 1] per-row scale
  float*       out = (float*)d_out;          // [B*S, 1024]

  const int n_tokens = in_sizes[0];
  const int blocks   = (n_tokens + TPB_TOKENS - 1) / TPB_TOKENS;

  qlv4_embed_dequant_kernel<<<blocks, THREADS, 0, stream>>>(ids, W, Sc, out,
                                                            n_tokens);
}